// MagNet_layer_27745488732707
// MI455X (gfx1250) — compile-verified
//
#include <hip/hip_runtime.h>

// Problem constants (match reference)
#define NN   1024
#define BB   8
#define FF   32
#define FOUTC 4

typedef __attribute__((ext_vector_type(2))) float v2f;
typedef __attribute__((ext_vector_type(8))) float v8f;

// ---------------------------------------------------------------------------
// Kernel 0: dinv[b,i] = D>0 ? 1/sqrt(D) : 0,  D = sum_j 0.5*(adj[b,i,j]+adj[b,j,i])
// One wave32 per row.
// ---------------------------------------------------------------------------
__global__ void magnet_dinv_kernel(const float* __restrict__ adj,
                                   float* __restrict__ dinv) {
  int wid  = (blockIdx.x * blockDim.x + threadIdx.x) >> 5;  // global wave id = b*NN + i
  int lane = threadIdx.x & 31;
  int b = wid >> 10;
  int i = wid & (NN - 1);
  const float* adjB = adj + (size_t)b * NN * NN;
  float s = 0.0f;
  for (int j = lane; j < NN; j += 32)
    s += 0.5f * (adjB[(size_t)i * NN + j] + adjB[(size_t)j * NN + i]);
  for (int off = 16; off > 0; off >>= 1)
    s += __shfl_xor(s, off, 32);
  if (lane == 0)
    dinv[wid] = (s > 0.0f) ? rsqrtf(s) : 0.0f;
}

// ---------------------------------------------------------------------------
// Kernel 1/2: Chebyshev pass.  Out = alpha * (L @ Z) + X   (complex)
//   L built on the fly from adj/dinv; Z is X (pass A) or z1 (pass B).
// grid = (NN/64, BB, FOUTC), block = 128 (4 waves, each owns 16 rows x 32 cols)
// ---------------------------------------------------------------------------
__global__ __launch_bounds__(128)
void magnet_cheb_kernel(const float* __restrict__ adj,
                        const float* __restrict__ dinv,
                        const float* __restrict__ Zr, const float* __restrict__ Zi,
                        const float* __restrict__ Xr, const float* __restrict__ Xi,
                        float* __restrict__ Or, float* __restrict__ Oi,
                        float alpha, int zHasIdx) {
  const int lane = threadIdx.x & 31;
  const int wave = threadIdx.x >> 5;
  const int b    = blockIdx.y;
  const int idx  = blockIdx.z;
  const int rows0 = (blockIdx.x * 4 + wave) * 16;

  // q = 1/FREQS[idx], FREQS = {2,3,4,5};  theta scale = 2*pi*q
  const float invf = (idx == 0) ? 0.5f : (idx == 1) ? (1.0f / 3.0f)
                    : (idx == 2) ? 0.25f : 0.2f;
  const float q2pi = 6.28318530717958647692f * invf;

  // A-matrix (16x4 f32) lane mapping: lanes 0-15 -> M, K={0,1}; lanes 16-31 -> M, K={2,3}
  const int i    = rows0 + (lane & 15);
  const int koff = (lane >> 4) << 1;   // 0 or 2
  const int f0   = lane & 15;          // B/D column within 16-wide tile

  const float* adjB = adj + (size_t)b * NN * NN;
  const float* dv   = dinv + b * NN;
  const float  di   = dv[i];

  const size_t zbase = ((size_t)((zHasIdx ? idx * BB : 0) + b)) * NN * FF;
  const size_t xbase = (size_t)b * NN * FF;
  const size_t obase = ((size_t)(idx * BB + b)) * NN * FF;

  v8f hr0 = {0.f,0.f,0.f,0.f,0.f,0.f,0.f,0.f};
  v8f hr1 = hr0, hi0 = hr0, hi1 = hr0;

  for (int k0 = 0; k0 < NN; k0 += 4) {
    const int j0 = k0 + koff, j1 = j0 + 1;

    // Build L tile elements on the fly (adj stays hot in the 192MB L2)
    float aij0 = adjB[(size_t)i  * NN + j0];
    float aji0 = adjB[(size_t)j0 * NN + i ];
    float aij1 = adjB[(size_t)i  * NN + j1];
    float aji1 = adjB[(size_t)j1 * NN + i ];
    float s0 = 0.5f * (aij0 + aji0) * di * dv[j0];
    float s1 = 0.5f * (aij1 + aji1) * di * dv[j1];
    float sn0, c0, sn1, c1;
    __sincosf(q2pi * (aij0 - aji0), &sn0, &c0);
    __sincosf(q2pi * (aij1 - aji1), &sn1, &c1);
    float lr0 = ((i == j0) ? 1.0f : 0.0f) - s0 * c0;
    float lr1 = ((i == j1) ? 1.0f : 0.0f) - s1 * c1;
    float li0 = -s0 * sn0;
    float li1 = -s1 * sn1;

    v2f Ar  = { lr0,  lr1 };
    v2f Ai  = { li0,  li1 };
    v2f AiN = { -li0, -li1 };   // f32 WMMA has no A/B NEG -> pre-negate in VALU

    // B tiles (4x16 f32): VGPR0 holds row K=koff, VGPR1 holds row K=koff+1
    const float* zr0 = Zr + zbase + (size_t)j0 * FF;
    const float* zr1 = Zr + zbase + (size_t)j1 * FF;
    const float* zi0 = Zi + zbase + (size_t)j0 * FF;
    const float* zi1 = Zi + zbase + (size_t)j1 * FF;
    v2f Br0 = { zr0[f0],      zr1[f0]      };
    v2f Br1 = { zr0[f0 + 16], zr1[f0 + 16] };
    v2f Bi0 = { zi0[f0],      zi1[f0]      };
    v2f Bi1 = { zi0[f0 + 16], zi1[f0 + 16] };

    // Complex accumulate: Hr += Ar*Br - Ai*Bi ; Hi += Ai*Br + Ar*Bi
    hr0 = __builtin_amdgcn_wmma_f32_16x16x4_f32(false, Ar,  false, Br0, (short)0, hr0, false, false);
    hr0 = __builtin_amdgcn_wmma_f32_16x16x4_f32(false, AiN, false, Bi0, (short)0, hr0, false, false);
    hr1 = __builtin_amdgcn_wmma_f32_16x16x4_f32(false, Ar,  false, Br1, (short)0, hr1, false, false);
    hr1 = __builtin_amdgcn_wmma_f32_16x16x4_f32(false, AiN, false, Bi1, (short)0, hr1, false, false);
    hi0 = __builtin_amdgcn_wmma_f32_16x16x4_f32(false, Ai,  false, Br0, (short)0, hi0, false, false);
    hi0 = __builtin_amdgcn_wmma_f32_16x16x4_f32(false, Ar,  false, Bi0, (short)0, hi0, false, false);
    hi1 = __builtin_amdgcn_wmma_f32_16x16x4_f32(false, Ai,  false, Br1, (short)0, hi1, false, false);
    hi1 = __builtin_amdgcn_wmma_f32_16x16x4_f32(false, Ar,  false, Bi1, (short)0, hi1, false, false);
  }

  // D tile (16x16 f32): VGPR v -> row rows0 + v (+8 for lanes 16-31), col = f0
  const int rbase = rows0 + ((lane >> 4) << 3);
#pragma unroll
  for (int v = 0; v < 8; ++v) {
    const int row = rbase + v;
    const size_t xo = xbase + (size_t)row * FF + f0;
    const size_t oo = obase + (size_t)row * FF + f0;
    Or[oo]      = alpha * hr0[v] + Xr[xo];
    Or[oo + 16] = alpha * hr1[v] + Xr[xo + 16];
    Oi[oo]      = alpha * hi0[v] + Xi[xo];
    Oi[oo + 16] = alpha * hi1[v] + Xi[xo + 16];
  }
}

// ---------------------------------------------------------------------------
// Kernel 3: projection.  out_r[b,n,idx] = sum_f X*w0 + z1*w1 + z2*w2 + bias
// ---------------------------------------------------------------------------
__global__ void magnet_proj_kernel(const float* __restrict__ Xr, const float* __restrict__ Xi,
                                   const float* __restrict__ z1r, const float* __restrict__ z1i,
                                   const float* __restrict__ z2r, const float* __restrict__ z2i,
                                   const float* __restrict__ weight,
                                   const float* __restrict__ bias,
                                   float* __restrict__ out) {
  int t = blockIdx.x * blockDim.x + threadIdx.x;        // over B*N*FOUT
  int idx = t & 3;
  int n   = (t >> 2) & (NN - 1);
  int b   = t >> 12;
  const float* w = weight + idx * (3 * FF);             // [k][f], K+1 = 3
  const size_t xo = ((size_t)b * NN + n) * FF;
  const size_t zo = (((size_t)idx * BB + b) * NN + n) * FF;
  float sr = 0.0f, si = 0.0f;
#pragma unroll 8
  for (int f = 0; f < FF; ++f) {
    float w0 = w[f], w1 = w[FF + f], w2 = w[2 * FF + f];
    sr += Xr[xo + f] * w0 + z1r[zo + f] * w1 + z2r[zo + f] * w2;
    si += Xi[xo + f] * w0 + z1i[zo + f] * w1 + z2i[zo + f] * w2;
  }
  out[t] = sr + bias[idx];                 // real block: [b][n][idx]
  out[(size_t)BB * NN * FOUTC + t] = si;   // imag block
}

// ---------------------------------------------------------------------------
extern "C" void kernel_launch(void* const* d_in, const int* in_sizes, int n_in,
                              void* d_out, int out_size, void* d_ws, size_t ws_size,
                              hipStream_t stream) {
  const float* Xr     = (const float*)d_in[0];
  const float* Xi     = (const float*)d_in[1];
  const float* adj    = (const float*)d_in[2];
  const float* weight = (const float*)d_in[3];
  const float* bias   = (const float*)d_in[4];
  float* out = (float*)d_out;

  // Workspace layout: dinv (32KB, padded to 64KB) | z1r | z1i | z2r | z2i (4MB each)
  char* ws = (char*)d_ws;
  float* dinv = (float*)ws;
  const size_t zElems = (size_t)FOUTC * BB * NN * FF;   // 1,048,576 floats
  float* z1r = (float*)(ws + (1 << 16));
  float* z1i = z1r + zElems;
  float* z2r = z1i + zElems;
  float* z2i = z2r + zElems;

  // 1) degree^{-1/2}
  magnet_dinv_kernel<<<dim3((BB * NN) / 8), 256, 0, stream>>>(adj, dinv);

  // 2) z1 = X - L X ; 3) z2 = 2 L z1 + X   (stream-ordered)
  dim3 grid(NN / 64, BB, FOUTC);
  magnet_cheb_kernel<<<grid, 128, 0, stream>>>(adj, dinv, Xr, Xi, Xr, Xi,
                                               z1r, z1i, -1.0f, 0);
  magnet_cheb_kernel<<<grid, 128, 0, stream>>>(adj, dinv, z1r, z1i, Xr, Xi,
                                               z2r, z2i, 2.0f, 1);

  // 4) channel projection + bias
  magnet_proj_kernel<<<dim3((BB * NN * FOUTC) / 256), 256, 0, stream>>>(
      Xr, Xi, z1r, z1i, z2r, z2i, weight, bias, out);
}